// PointTriplaneGenerator_42666205118756
// MI455X (gfx1250) — compile-verified
//
#include <hip/hip_runtime.h>
#include <hip/hip_bf16.h>

#define Gg   128
#define Cc   256
#define Hh   4
#define Dd   64
#define G2   16384
#define LN_EPS 1e-5f

typedef float v2f __attribute__((ext_vector_type(2)));
typedef float v8f __attribute__((ext_vector_type(8)));

__device__ __forceinline__ void atomicMaxFloat(float* addr, float val) {
    // monotonic-bits trick: works with -inf init (0xFF800000)
    if (val >= 0.0f) atomicMax((int*)addr, __float_as_int(val));
    else             atomicMin((unsigned int*)addr, (unsigned int)__float_as_int(val));
}

// ---------------------------------------------------------------- fill bits
__global__ void fill_bits_kernel(unsigned int* p, long long n, unsigned int bits) {
    long long i = (long long)blockIdx.x * blockDim.x + threadIdx.x;
    long long stride = (long long)gridDim.x * blockDim.x;
    for (; i < n; i += stride) p[i] = bits;
}

// ------------------------------------------------- q transpose: (C,G2)->(G2,C)
__global__ void qt_kernel(const float* __restrict__ qxy,
                          const float* __restrict__ qxz,
                          const float* __restrict__ qyz,
                          float* __restrict__ qT) {
    const float* q = (blockIdx.z == 0) ? qxy : (blockIdx.z == 1 ? qxz : qyz);
    __shared__ float tile[32][33];
    const int cell0 = blockIdx.x * 32;
    const int c0    = blockIdx.y * 32;
    const int tx = threadIdx.x, ty = threadIdx.y;   // 32 x 8
    #pragma unroll
    for (int r = 0; r < 32; r += 8)
        tile[ty + r][tx] = q[(size_t)(c0 + ty + r) * G2 + cell0 + tx];
    __syncthreads();
    float* out = qT + (size_t)blockIdx.z * G2 * Cc;
    #pragma unroll
    for (int r = 0; r < 32; r += 8)
        out[(size_t)(cell0 + ty + r) * Cc + c0 + tx] = tile[tx][ty + r];
}

// ---------------------------------------------------------------- shared GEMM helper
// Block = 256 threads = 8 waves; computes LN(feats) for a 16-point tile into fnL,
// then OUT = fnL x W^T + bias via V_WMMA_F32_16X16X4_F32 into outL[16][256].
struct TileCtx {
    long long gp;     // global point id (b*N + n)
    bool valid;
    int p, cs, wave, lane, hf, ln16;
};

__device__ __forceinline__ void tile_ln_gemm(
    const float* __restrict__ pts, const float* __restrict__ bounds,
    const float* __restrict__ W, const float* __restrict__ bias,
    const float* __restrict__ lng, const float* __restrict__ lnb,
    long long BN, TileCtx& cx,
    float (*fnL)[Cc], float (*outL)[Cc],
    float (*redS)[16], float (*redQ)[16],
    float* muL, float* rsL, int (*cellL)[3], int* validL)
{
    const int t = threadIdx.x;
    cx.p    = t & 15;
    cx.cs   = t >> 4;
    cx.wave = t >> 5;
    cx.lane = t & 31;
    cx.hf   = cx.lane >> 4;
    cx.ln16 = cx.lane & 15;
    cx.gp   = (long long)blockIdx.x * 16 + cx.p;
    cx.valid = (cx.gp < BN);

    float coord[3] = {0.f, 0.f, 0.f};
    if (cx.cs == 0) {
        validL[cx.p] = cx.valid ? 1 : 0;
        if (cx.valid) {
            const float* pt = pts + cx.gp * (size_t)Cc;
            #pragma unroll
            for (int i = 0; i < 3; ++i) {
                float lo = bounds[2 * i], hi = bounds[2 * i + 1];
                coord[i] = 2.0f * (pt[i] - lo) / (hi - lo) - 1.0f;
            }
        }
        const int a0[3] = {0, 0, 1}, a1[3] = {1, 2, 2};
        #pragma unroll
        for (int pl = 0; pl < 3; ++pl) {
            int g0 = (int)((coord[a0[pl]] * 0.5f + 0.5f) * (float)(Gg - 1));
            int g1 = (int)((coord[a1[pl]] * 0.5f + 0.5f) * (float)(Gg - 1));
            g0 = min(max(g0, 0), Gg - 1);
            g1 = min(max(g1, 0), Gg - 1);
            cellL[cx.p][pl] = g0 * Gg + g1;
        }
    }

    // load 16 channels per thread via float4 (64B aligned), splice coords into c<3
    float x[16];
    if (cx.valid) {
        const float4* p4 = (const float4*)(pts + cx.gp * (size_t)Cc + cx.cs * 16);
        #pragma unroll
        for (int q = 0; q < 4; ++q) {
            float4 f = p4[q];
            x[q * 4 + 0] = f.x; x[q * 4 + 1] = f.y;
            x[q * 4 + 2] = f.z; x[q * 4 + 3] = f.w;
        }
        if (cx.cs == 0) { x[0] = coord[0]; x[1] = coord[1]; x[2] = coord[2]; }
    } else {
        #pragma unroll
        for (int j = 0; j < 16; ++j) x[j] = 0.0f;
    }

    // two-pass LayerNorm statistics (matches reference numerics)
    float s1 = 0.f;
    #pragma unroll
    for (int j = 0; j < 16; ++j) s1 += x[j];
    redS[cx.p][cx.cs] = s1;
    __syncthreads();
    if (t < 16) {
        float a = 0.f;
        #pragma unroll
        for (int i = 0; i < 16; ++i) a += redS[t][i];
        muL[t] = a * (1.0f / Cc);
    }
    __syncthreads();
    {
        const float mu = muL[cx.p];
        float s2 = 0.f;
        #pragma unroll
        for (int j = 0; j < 16; ++j) { float d = x[j] - mu; s2 += d * d; }
        redQ[cx.p][cx.cs] = s2;
    }
    __syncthreads();
    if (t < 16) {
        float q = 0.f;
        #pragma unroll
        for (int i = 0; i < 16; ++i) q += redQ[t][i];
        rsL[t] = rsqrtf(q * (1.0f / Cc) + LN_EPS);
    }
    __syncthreads();
    {
        const float mu = muL[cx.p], rs = rsL[cx.p];
        #pragma unroll
        for (int j = 0; j < 16; ++j) {
            int c = cx.cs * 16 + j;
            fnL[cx.p][c] = (x[j] - mu) * rs * lng[c] + lnb[c];
        }
    }
    __syncthreads();

    // GEMM: each wave computes 2 column tiles of 16 output channels
    #pragma unroll
    for (int it = 0; it < 2; ++it) {
        const int nt = cx.wave * 2 + it;
        const int n  = nt * 16 + cx.ln16;
        const float* wrow = W + (size_t)n * Cc;   // B[k][n] = W[n][k], streams along k
        __builtin_prefetch(wrow, 0, 0);           // global_prefetch_b8
        __builtin_prefetch(wrow + 128, 0, 0);
        v8f acc = {};
        for (int k0 = 0; k0 < Cc; k0 += 4) {
            v2f a, b;
            a.x = fnL[cx.ln16][k0 + 2 * cx.hf];
            a.y = fnL[cx.ln16][k0 + 2 * cx.hf + 1];
            b.x = wrow[k0 + 2 * cx.hf];
            b.y = wrow[k0 + 2 * cx.hf + 1];
            acc = __builtin_amdgcn_wmma_f32_16x16x4_f32(false, a, false, b,
                                                        (short)0, acc, false, false);
        }
        const float bv = bias[n];
        #pragma unroll
        for (int r = 0; r < 8; ++r) {
            int m = r + cx.hf * 8;               // C/D layout: lanes16-31 hold M=8..15
            outL[m][n] = acc[r] + bv;
        }
    }
    __syncthreads();
}

// ---------------------------------------------------------------- pass 1: K, s, smax
__global__ __launch_bounds__(256) void pass1_kernel(
    const float* __restrict__ pts, const float* __restrict__ bounds,
    const float* __restrict__ qT,
    const float* __restrict__ Wk, const float* __restrict__ bk,
    const float* __restrict__ lng, const float* __restrict__ lnb,
    float* __restrict__ sBuf, float* __restrict__ smax,
    long long BN, long long Nn)
{
    __shared__ float fnL[16][Cc];
    __shared__ float ktL[16][Cc];
    __shared__ float redS[16][16], redQ[16][16];
    __shared__ float muL[16], rsL[16];
    __shared__ int   cellL[16][3];
    __shared__ int   validL[16];

    TileCtx cx;
    tile_ln_gemm(pts, bounds, Wk, bk, lng, lnb, BN, cx,
                 fnL, ktL, redS, redQ, muL, rsL, cellL, validL);

    const int t = threadIdx.x;
    if (t < 192) {
        const int pp   = t & 15;
        const int rest = t >> 4;          // 0..11
        const int h    = rest & 3;
        const int pl   = rest >> 2;
        if (validL[pp]) {
            long long gp = (long long)blockIdx.x * 16 + pp;
            long long b  = gp / Nn;
            long long n  = gp - b * Nn;
            const int cell = cellL[pp][pl];
            // qrow is 256B-aligned in ws -> float4 loads (global_load_b128)
            const float4* q4 = (const float4*)(qT + ((size_t)pl * G2 + cell) * Cc + h * Dd);
            float s = 0.0f;
            #pragma unroll
            for (int d4 = 0; d4 < Dd / 4; ++d4) {
                float4 qv = q4[d4];
                const float* kk = &ktL[pp][h * Dd + d4 * 4];
                s += kk[0] * qv.x + kk[1] * qv.y + kk[2] * qv.z + kk[3] * qv.w;
            }
            sBuf[(((long long)b * 3 + pl) * Nn + n) * Hh + h] = s;
            atomicMaxFloat(&smax[(((long long)b * 3 + pl) * G2 + cell) * Hh + h], s);
        }
    }
}

// ---------------------------------------------------------------- pass 2: V, e*v scatter
__global__ __launch_bounds__(256) void pass2_kernel(
    const float* __restrict__ pts, const float* __restrict__ bounds,
    const float* __restrict__ Wv, const float* __restrict__ bv,
    const float* __restrict__ lng, const float* __restrict__ lnb,
    const float* __restrict__ sBuf, const float* __restrict__ smax,
    float* __restrict__ den, float* __restrict__ out,
    long long BN, long long Nn)
{
    __shared__ float fnL[16][Cc];
    __shared__ float vtL[16][Cc];
    __shared__ float redS[16][16], redQ[16][16];
    __shared__ float muL[16], rsL[16];
    __shared__ int   cellL[16][3];
    __shared__ int   validL[16];
    __shared__ float eL[16][12];         // [point][pl*4+h]

    TileCtx cx;
    tile_ln_gemm(pts, bounds, Wv, bv, lng, lnb, BN, cx,
                 fnL, vtL, redS, redQ, muL, rsL, cellL, validL);

    const int t = threadIdx.x;
    if (t < 192) {
        const int pp   = t & 15;
        const int rest = t >> 4;
        const int h    = rest & 3;
        const int pl   = rest >> 2;
        if (validL[pp]) {
            long long gp = (long long)blockIdx.x * 16 + pp;
            long long b  = gp / Nn;
            long long n  = gp - b * Nn;
            const int cell = cellL[pp][pl];
            float s  = sBuf[(((long long)b * 3 + pl) * Nn + n) * Hh + h];
            float sm = smax[(((long long)b * 3 + pl) * G2 + cell) * Hh + h];
            float e  = __expf(s - sm);
            eL[pp][pl * 4 + h] = e;
            atomicAdd(&den[(((long long)b * 3 + pl) * G2 + cell) * Hh + h], e);
        }
    }
    __syncthreads();

    if (validL[cx.p]) {
        long long b = cx.gp / Nn;
        const int h = cx.cs >> 2;        // channels cs*16.. stay in one head block
        #pragma unroll
        for (int pl = 0; pl < 3; ++pl) {
            const int cell = cellL[cx.p][pl];
            const float e  = eL[cx.p][pl * 4 + h];
            float* o = out + (((long long)b * 3 + pl) * G2 + cell) * Cc + cx.cs * 16;
            #pragma unroll
            for (int j = 0; j < 16; ++j)
                atomicAdd(&o[j], e * vtL[cx.p][cx.cs * 16 + j]);
        }
    }
}

// ---------------------------------------------------------------- pass 3: normalize
__global__ __launch_bounds__(256) void normalize_kernel(
    float* __restrict__ out, const float* __restrict__ den, long long total)
{
    long long i = (long long)blockIdx.x * blockDim.x + threadIdx.x;
    long long stride = (long long)gridDim.x * blockDim.x;
    for (; i < total; i += stride) {
        int c = (int)(i & (Cc - 1));
        long long cellIdx = i >> 8;
        float d = den[cellIdx * Hh + (c >> 6)];
        out[i] = out[i] / fmaxf(d, 1e-30f);
    }
}

// ---------------------------------------------------------------- launch
extern "C" void kernel_launch(void* const* d_in, const int* in_sizes, int n_in,
                              void* d_out, int out_size, void* d_ws, size_t ws_size,
                              hipStream_t stream) {
    (void)n_in; (void)ws_size;
    const float* pts    = (const float*)d_in[0];
    const float* bounds = (const float*)d_in[1];
    const float* qxy    = (const float*)d_in[2];
    const float* qxz    = (const float*)d_in[3];
    const float* qyz    = (const float*)d_in[4];
    const float* Wk     = (const float*)d_in[5];
    const float* bk     = (const float*)d_in[6];
    const float* Wv     = (const float*)d_in[7];
    const float* bv     = (const float*)d_in[8];
    const float* lng    = (const float*)d_in[9];
    const float* lnb    = (const float*)d_in[10];
    float* out = (float*)d_out;

    const long long BN = (long long)in_sizes[0] / Cc;           // B*N
    const long long Bb = (long long)out_size / (3LL * Cc * G2); // batches
    const long long Nn = BN / Bb;                               // points per batch

    // workspace layout (floats)
    float* ws   = (float*)d_ws;
    float* qT   = ws;                       size_t off = 3ULL * G2 * Cc;
    float* sBuf = ws + off;                 off += (size_t)Bb * 3 * Nn * Hh;
    float* smax = ws + off;                 off += (size_t)Bb * 3 * G2 * Hh;
    float* den  = ws + off;                 off += (size_t)Bb * 3 * G2 * Hh;

    hipMemsetAsync(d_out, 0, (size_t)out_size * sizeof(float), stream);
    hipMemsetAsync(den, 0, (size_t)Bb * 3 * G2 * Hh * sizeof(float), stream);

    {
        long long n = (long long)Bb * 3 * G2 * Hh;
        fill_bits_kernel<<<256, 256, 0, stream>>>((unsigned int*)smax, n, 0xFF800000u); // -inf
    }

    qt_kernel<<<dim3(G2 / 32, Cc / 32, 3), dim3(32, 8), 0, stream>>>(qxy, qxz, qyz, qT);

    const unsigned int ntiles = (unsigned int)((BN + 15) / 16);
    pass1_kernel<<<ntiles, 256, 0, stream>>>(pts, bounds, qT, Wk, bk, lng, lnb,
                                             sBuf, smax, BN, Nn);
    pass2_kernel<<<ntiles, 256, 0, stream>>>(pts, bounds, Wv, bv, lng, lnb,
                                             sBuf, smax, den, out, BN, Nn);

    const long long total = (long long)Bb * 3 * G2 * Cc;
    normalize_kernel<<<(unsigned int)((total + 255) / 256), 256, 0, stream>>>(out, den, total);
}